// LieGroupDecoder_18751827214622
// MI455X (gfx1250) — compile-verified
//
#include <hip/hip_runtime.h>
#include <hip/hip_bf16.h>

// ---------------------------------------------------------------------------
// LieGroupDecoder on MI455X (gfx1250, wave32, WMMA + async VMEM->LDS).
//
// Sizes: B=2048, NUM_WS=16, Z=512, MAT=10 (MM=100), C_mid=C_out=512, L=16.
// Heavy math runs through V_WMMA_F32_16X16X4_F32.  The workload is
// memory-bound (~0.7 GB activation traffic @ 23.3 TB/s vs ~216 GFLOP), so
// fp32 matrix ops cost nothing; L2 weight traffic is cut 4x by batching
// NB=4 batch elements per block, and activation staging uses the gfx1250
// async VMEM->LDS path (GLOBAL_LOAD_ASYNC_TO_LDS_B32, ASYNCcnt).
// ---------------------------------------------------------------------------

typedef __attribute__((ext_vector_type(2))) float v2f;
typedef __attribute__((ext_vector_type(8))) float v8f;

__device__ __forceinline__ v8f wmma_f32_16x16x4(v2f a, v2f b, v8f c) {
  // (neg_a, A, neg_b, B, c_mod, C, reuse_a, reuse_b)
  return __builtin_amdgcn_wmma_f32_16x16x4_f32(false, a, false, b,
                                               (short)0, c, false, false);
}

// Async copy global -> LDS (no VGPR staging), tracked by ASYNCcnt.
// VDST vgpr carries the 32-bit LDS byte offset; a generic LDS address
// truncates to addr[31:0] == LDS offset (ISA 10.2), so the cast is exact.
__device__ __forceinline__ void async_ld_b32(unsigned lds_byte, const float* g) {
  asm volatile("global_load_async_to_lds_b32 %0, %1, off"
               :: "v"(lds_byte), "v"(g) : "memory");
}
__device__ __forceinline__ void wait_asynccnt0() {
  asm volatile("s_wait_asynccnt 0x0" ::: "memory");
}

// Fragment layout (ISA 7.12.2, wave32):
//   A (16x4 f32, 2 VGPRs): lane%16 = M row, K = 2*(lane>>4) + vgpr
//   B (4x16 f32, 2 VGPRs): lane%16 = N col, K = 2*(lane>>4) + vgpr
//   C/D (16x16, 8 VGPRs):  M = r + 8*(lane>>4), N = lane%16

// ---------------------------------------------------------------------------
// Kernel 1: lie_alg[b,w,ij] = sum_z z[b,w,z] * basis[w,z,ij]
// Per w: GEMM M=2048(b) x N=100(ij) x K=512.  One wave per 16x16 tile.
// Out-of-range columns (n in [100,112)) use a clamped address + cndmask-0,
// keeping all loads unconditional (no EXEC branching in the hot loop).
// ---------------------------------------------------------------------------
__global__ __launch_bounds__(32)
void lie_gemm_kernel(const float* __restrict__ z,
                     const float* __restrict__ basis,
                     float* __restrict__ lie)
{
  const int lane  = threadIdx.x;
  const int laneN = lane & 15;
  const int hi    = lane >> 4;
  const int b0    = blockIdx.x * 16;
  const int n0    = blockIdx.y * 16;
  const int w     = blockIdx.z;
  const int n     = n0 + laneN;
  const bool nok  = (n < 100);
  const int nc    = nok ? n : 99;              // clamped, always in-bounds

  // A rows: z[(b0+laneN), w, k]  (contiguous pair at k = k0 + 2*hi)
  const float* zrow = z + ((size_t)(b0 + laneN) * 16 + w) * 512 + 2 * hi;
  // B rows: basis[w, k, nc]
  const float* bcol = basis + (size_t)w * 512 * 100 + nc;

  v8f c = {};
  for (int k0 = 0; k0 < 512; k0 += 4) {
    v2f a = *(const v2f*)(zrow + k0);
    const int kidx = k0 + 2 * hi;
    const float t0 = bcol[(size_t)kidx * 100];        // unconditional load
    const float t1 = bcol[(size_t)(kidx + 1) * 100];
    v2f bv;
    bv.x = nok ? t0 : 0.f;                            // v_cndmask
    bv.y = nok ? t1 : 0.f;
    c = wmma_f32_16x16x4(a, bv, c);
  }

  if (nok) {
#pragma unroll
    for (int r = 0; r < 8; ++r) {
      const int bb = b0 + r + 8 * hi;
      lie[((size_t)bb * 16 + w) * 100 + n] = c[r];
    }
  }
}

// ---------------------------------------------------------------------------
// Kernel 2: expm of 10x10 matrices via Taylor-9 (||A|| ~ 0.2 => exact to
// fp32 ulp).  One 128-thread block per matrix, thread t owns element
// (t/10, t%10); ping-pong term buffers in LDS.
// Output written pre-transposed as gfeat[b, c, w] so conv layer 0 can
// consume it as an NCH tensor directly.
// ---------------------------------------------------------------------------
__global__ __launch_bounds__(128)
void expm10_kernel(const float* __restrict__ lie, float* __restrict__ gfeat)
{
  __shared__ float Ash[100];
  __shared__ float Tsh[2][100];

  const int t   = threadIdx.x;
  const int mat = blockIdx.x;          // = b*16 + w
  const int b   = mat >> 4;
  const int w   = mat & 15;

  float p = 0.f;
  int i = 0, j = 0;
  if (t < 100) {
    Ash[t] = lie[(size_t)mat * 100 + t];
    i = t / 10;
    j = t - i * 10;
    p = (i == j) ? 1.f : 0.f;          // P = I, term T0 = I
    Tsh[0][t] = p;
  }
  __syncthreads();

  int cur = 0;
  for (int k = 1; k <= 9; ++k) {
    if (t < 100) {
      float tn = 0.f;
#pragma unroll
      for (int l = 0; l < 10; ++l)
        tn += Ash[i * 10 + l] * Tsh[cur][l * 10 + j];
      tn *= 1.f / (float)k;            // T_k = A*T_{k-1}/k
      Tsh[1 - cur][t] = tn;
      p += tn;
    }
    cur = 1 - cur;
    __syncthreads();
  }

  if (t < 100)
    gfeat[((size_t)b * 100 + t) * 16 + w] = p;
}

// ---------------------------------------------------------------------------
// Kernel 3: conv1d (K=3, pad=1) over L=16 as a WMMA GEMM.
//   out[b,co,l] = bias[co] + sum_{ci,k} W[co,ci,k] * x[b,ci,l+k-1]
// K-dim ordering kidx = ci*3+k matches W's [co][ci][3] row-major layout, so
// A-fragments are contiguous 64-bit loads.
//
// Tiling: one block = 512 threads (16 waves) covers NB=4 batch elements and
// all 32 channel tiles (each wave does 2 tiles).  Each weight A-fragment is
// reused for 4 WMMAs (one per batch element), cutting L2 weight traffic 4x.
// x is staged in dynamic LDS with a zero halo via async VMEM->LDS copies:
// xp[nb][ci][18], xp[nb][ci][1+l] = x[b0+nb,ci,l], so B elements are
// xp[nb][ci][l+k].  Bias folded into the accumulator init; ReLU fused at
// store.  `transposed` stores [b,l,co] for the final layer (safe in place:
// all global reads of a block land in LDS before the barrier; stores only
// touch its own batches).
// ---------------------------------------------------------------------------
#define CONV_NB 4

__global__ __launch_bounds__(512)
void conv1d_wmma_kernel(const float* xin,
                        const float* __restrict__ Wt,
                        const float* __restrict__ bias,
                        float* xout,
                        int Cin, int relu, int transposed)
{
  extern __shared__ float xp[];        // CONV_NB * Cin * 18 floats

  const int tid = threadIdx.x;
  const int b0  = blockIdx.x * CONV_NB;
  const int K   = Cin * 3;
  const int CH  = Cin * 18;            // LDS floats per batch element

  // Stage x[b0..b0+3] into LDS (async, no VGPR staging) + zero halos.
#pragma unroll
  for (int nb = 0; nb < CONV_NB; ++nb) {
    const float* xb  = xin + (size_t)(b0 + nb) * Cin * 16;
    float*       xpn = xp + nb * CH;
    for (int r = tid; r < Cin * 16; r += 512) {
      const int ci = r >> 4, l = r & 15;
      async_ld_b32((unsigned)(uintptr_t)(xpn + ci * 18 + 1 + l), xb + r);
    }
    for (int ci = tid; ci < Cin; ci += 512) {
      xpn[ci * 18]      = 0.f;
      xpn[ci * 18 + 17] = 0.f;
    }
  }
  wait_asynccnt0();                    // this wave's async copies landed
  __syncthreads();                     // make all waves' copies visible

  const int wave  = tid >> 5;          // 0..15
  const int lane  = tid & 31;
  const int laneN = lane & 15;         // N = output position l
  const int hi    = lane >> 4;

  for (int half = 0; half < 2; ++half) {
    const int co0 = (wave + half * 16) * 16;   // channel tile base

    v8f c[CONV_NB];
#pragma unroll
    for (int nb = 0; nb < CONV_NB; ++nb)
#pragma unroll
      for (int r = 0; r < 8; ++r) c[nb][r] = bias[co0 + r + 8 * hi];

    const float* wrow = Wt + (size_t)(co0 + laneN) * K + 2 * hi;

#pragma unroll 4
    for (int k0 = 0; k0 < K; k0 += 4) {
      v2f a = *(const v2f*)(wrow + k0);             // W[co, kidx..kidx+1]
      const int kidx = k0 + 2 * hi;
      const int ci0 = kidx / 3,       kk0 = kidx - ci0 * 3;
      const int ci1 = (kidx + 1) / 3, kk1 = (kidx + 1) - ci1 * 3;
      const int o0 = ci0 * 18 + kk0 + laneN;
      const int o1 = ci1 * 18 + kk1 + laneN;
#pragma unroll
      for (int nb = 0; nb < CONV_NB; ++nb) {        // reuse A across batches
        v2f bv;
        bv.x = xp[nb * CH + o0];
        bv.y = xp[nb * CH + o1];
        c[nb] = wmma_f32_16x16x4(a, bv, c[nb]);
      }
    }

#pragma unroll
    for (int nb = 0; nb < CONV_NB; ++nb) {
#pragma unroll
      for (int r = 0; r < 8; ++r) {
        const int co = co0 + r + 8 * hi;
        float v = c[nb][r];
        if (relu) v = fmaxf(v, 0.f);
        const size_t idx = transposed
            ? ((size_t)(b0 + nb) * 16 + laneN) * 512 + co   // [b, l, co]
            : ((size_t)(b0 + nb) * 512 + co) * 16 + laneN;  // [b, co, l]
        xout[idx] = v;
      }
    }
  }
}

// ---------------------------------------------------------------------------
// Launch.  Workspace: gfeat (13.1 MB) + one 67 MB activation buffer (the lie
// buffer overlaps the activation buffer — dead once expm has run).  d_out
// doubles as the second ping-pong buffer; the last layer transposes in place.
// ---------------------------------------------------------------------------
extern "C" void kernel_launch(void* const* d_in, const int* in_sizes, int n_in,
                              void* d_out, int out_size, void* d_ws, size_t ws_size,
                              hipStream_t stream)
{
  (void)in_sizes; (void)n_in; (void)out_size; (void)ws_size;

  const float* z     = (const float*)d_in[0];
  const float* basis = (const float*)d_in[1];
  const float* W0 = (const float*)d_in[2];  const float* b0 = (const float*)d_in[3];
  const float* W1 = (const float*)d_in[4];  const float* b1 = (const float*)d_in[5];
  const float* W2 = (const float*)d_in[6];  const float* b2 = (const float*)d_in[7];
  const float* W3 = (const float*)d_in[8];  const float* b3 = (const float*)d_in[9];
  const float* W4 = (const float*)d_in[10]; const float* b4 = (const float*)d_in[11];
  float* out = (float*)d_out;

  char* ws = (char*)d_ws;
  const size_t GF_BYTES = (size_t)2048 * 100 * 16 * sizeof(float);  // 13.1 MB
  float* gfeat = (float*)ws;                // [b, 100, 16]
  float* lie   = (float*)(ws + GF_BYTES);   // [b, w, 100]   (dead after expm)
  float* act   = (float*)(ws + GF_BYTES);   // [b, 512, 16]  (67 MB, reuses lie)

  // 1) lie algebra combination (16 GEMMs, WMMA f32)
  lie_gemm_kernel<<<dim3(128, 7, 16), 32, 0, stream>>>(z, basis, lie);

  // 2) batched 10x10 expm, writes gfeat transposed
  expm10_kernel<<<dim3(2048 * 16), 128, 0, stream>>>(lie, gfeat);

  // 3) conv decoder (WMMA GEMMs); d_out used as ping-pong scratch
  const int    nblk    = 2048 / CONV_NB;                       // 512
  const size_t lds100  = (size_t)CONV_NB * 100 * 18 * 4;       // 28.8 KB
  const size_t lds512  = (size_t)CONV_NB * 512 * 18 * 4;       // 144 KB
  conv1d_wmma_kernel<<<nblk, 512, lds100, stream>>>(gfeat, W0, b0, act, 100, 1, 0);
  conv1d_wmma_kernel<<<nblk, 512, lds512, stream>>>(act,   W1, b1, out, 512, 1, 0);
  conv1d_wmma_kernel<<<nblk, 512, lds512, stream>>>(out,   W2, b2, act, 512, 1, 0);
  conv1d_wmma_kernel<<<nblk, 512, lds512, stream>>>(act,   W3, b3, out, 512, 1, 0);
  conv1d_wmma_kernel<<<nblk, 512, lds512, stream>>>(out,   W4, b4, out, 512, 0, 1);
}